// SymbolSubcarrierAutocorrelation_23484881174749
// MI455X (gfx1250) — compile-verified
//
#include <hip/hip_runtime.h>
#include <stdint.h>

typedef float v2f __attribute__((ext_vector_type(2)));
typedef float v8f __attribute__((ext_vector_type(8)));

namespace {
constexpr int NSYM = 5;              // symbols (S)
constexpr int NSUB = 33;             // subcarriers (W)
constexpr int NANT = 16;             // TXH*TXW (A)
constexpr int ROWS = NSYM * NANT;    // 80 rows, ordered (s, a) -> lr = s*16 + a
constexpr int KP   = 36;             // K padded to multiple of 4
constexpr int NP   = 48;             // N padded to 3 WMMA tiles
constexpr int KSTEPS = KP / 4;       // 9 WMMA k-steps
constexpr int NPAIR = KP / 2;        // packed (k, k+1) float2 rows of B
constexpr int THREADS = 160;         // 5 waves -> one 16-row block (one s) per wave
constexpr int NBLK = ROWS / 16;      // 5 (block rb <=> symbol s = rb)
constexpr int NWAVES = THREADS / 32;
constexpr int SLICE = NSYM * NANT * NSUB;   // 2640 floats per (n, component)

// LDS layout (float offsets)
constexpr int B1R = 0;                       // F_sub real, packed pairs   (1728)
constexpr int B1I = B1R + NPAIR * NP * 2;    // F_sub imag                 (1728)
constexpr int B2R = B1I + NPAIR * NP * 2;    // F_sub^H real               (1728)
constexpr int B2I = B2R + NPAIR * NP * 2;    // F_sub^H imag               (1728)
constexpr int FSR = B2I + NPAIR * NP * 2;    // F_sym real (25, pad to 32)
constexpr int FSI = FSR + 32;                // F_sym imag
constexpr int X0R = FSI + 32;                // working tile real (80x36)
constexpr int X0I = X0R + ROWS * KP;         // working tile imag
constexpr int PWR = X0I + ROWS * KP;         // |X|^2, real (80x36)
constexpr int LDS_FLOATS = PWR + ROWS * KP;  // 15616 floats = 62464 B
}

// CDNA5 async global->LDS (GVS form): LDS byte address in VGPR, 32-bit byte
// offset in VGPR, 64-bit base in SGPR pair. Tracked by ASYNCcnt.
__device__ __forceinline__ void async_g2l_b32(uint32_t lds_byte, uint32_t voff,
                                              const float* base) {
    asm volatile("global_load_async_to_lds_b32 %0, %1, %2"
                 :: "v"(lds_byte), "v"(voff), "s"(base) : "memory");
}
__device__ __forceinline__ void wait_asynccnt0() {
    asm volatile("s_wait_asynccnt 0x0" ::: "memory");
}

__global__ __launch_bounds__(THREADS)
void ssa_autocorr_kernel(const float* __restrict__ gxr, const float* __restrict__ gxi,
                         const float* __restrict__ gfsr, const float* __restrict__ gfsi,
                         const float* __restrict__ gfbr, const float* __restrict__ gfbi,
                         float* __restrict__ gout)
{
    __shared__ float lds[LDS_FLOATS];
    const int tid = threadIdx.x;
    const int n = blockIdx.x;                    // one batch element per workgroup
    const float* srcR = gxr + (long)n * SLICE;   // contiguous (s, a, w) slice
    const float* srcI = gxi + (long)n * SLICE;

    // ---- kick off async global->LDS of the whole input tile (ASYNCcnt path).
    //      LDS rows keep the global (s,a) order; only the 33->36 K pad is inserted.
    for (int i = tid; i < SLICE; i += THREADS) {
        const int lr = i / NSUB;
        const int w  = i - lr * NSUB;
        const uint32_t dR = (uint32_t)(uintptr_t)&lds[X0R + lr * KP + w];
        const uint32_t dI = (uint32_t)(uintptr_t)&lds[X0I + lr * KP + w];
        const uint32_t vo = (uint32_t)i * 4u;
        async_g2l_b32(dR, vo, srcR);
        async_g2l_b32(dI, vo, srcI);
    }
    // zero the K pad columns (w = 33..35) while the async loads are in flight
    for (int i = tid; i < ROWS * (KP - NSUB); i += THREADS) {
        const int lr = i / (KP - NSUB);
        const int p  = i - lr * (KP - NSUB);
        lds[X0R + lr * KP + NSUB + p] = 0.f;
        lds[X0I + lr * KP + NSUB + p] = 0.f;
    }

    // ---- F_sym coefficients (broadcast reads later are conflict-free)
    for (int i = tid; i < NSYM * NSYM; i += THREADS) {
        lds[FSR + i] = gfsr[i];
        lds[FSI + i] = gfsi[i];
    }

    // ---- zero-padded, K-pair-packed B operands: B1 = F_sub, B2 = F_sub^H
    for (int i = tid; i < NPAIR * NP; i += THREADS) {
        const int p = i / NP, v = i % NP;
        const int k0 = 2 * p, k1 = 2 * p + 1;
        const bool vok = v < NSUB;
        const bool k0ok = (k0 < NSUB) && vok;
        const bool k1ok = (k1 < NSUB) && vok;
        v2f fr, fi, tr, ti;
        fr.x = k0ok ? gfbr[k0 * NSUB + v] : 0.f;
        fr.y = k1ok ? gfbr[k1 * NSUB + v] : 0.f;
        fi.x = k0ok ? gfbi[k0 * NSUB + v] : 0.f;
        fi.y = k1ok ? gfbi[k1 * NSUB + v] : 0.f;
        tr.x = k0ok ? gfbr[v * NSUB + k0] : 0.f;
        tr.y = k1ok ? gfbr[v * NSUB + k1] : 0.f;
        ti.x = k0ok ? -gfbi[v * NSUB + k0] : 0.f;
        ti.y = k1ok ? -gfbi[v * NSUB + k1] : 0.f;
        ((v2f*)(lds + B1R))[i] = fr;
        ((v2f*)(lds + B1I))[i] = fi;
        ((v2f*)(lds + B2R))[i] = tr;
        ((v2f*)(lds + B2I))[i] = ti;
    }
    wait_asynccnt0();
    __syncthreads();

    // ---- forward symbol DFT (5x5 complex) per (a, w) column, in place.
    //      Rows for fixed (a,w) sit at stride 16*36 floats (s-major order).
    for (int c = tid; c < NANT * NSUB; c += THREADS) {
        const int a = c / NSUB, w = c - a * NSUB;
        const int base = a * KP + w;
        float xr[NSYM], xi[NSYM];
        #pragma unroll
        for (int s = 0; s < NSYM; ++s) {
            xr[s] = lds[X0R + base + s * NANT * KP];
            xi[s] = lds[X0I + base + s * NANT * KP];
        }
        #pragma unroll
        for (int s = 0; s < NSYM; ++s) {
            float ar = 0.f, ai = 0.f;
            #pragma unroll
            for (int t = 0; t < NSYM; ++t) {
                const float fr = lds[FSR + s * NSYM + t];
                const float fi = lds[FSI + s * NSYM + t];
                ar += fr * xr[t] - fi * xi[t];
                ai += fr * xi[t] + fi * xr[t];
            }
            lds[X0R + base + s * NANT * KP] = ar;
            lds[X0I + base + s * NANT * KP] = ai;
        }
    }
    __syncthreads();

    const int lane = tid & 31;
    const int wave = tid >> 5;
    const int half = lane >> 4;   // A/B fragments: lanes 16-31 hold K+2
    const int nl   = lane & 15;

    // ---- stage 1: X2 = X1 @ F_sub via 3-multiply complex WMMA, |X2|^2 -> PWR.
    //      A-fragments hoisted to registers once per row-block (reused by all
    //      3 N-tiles; avoids 3x redundant LDS reads the aliasing rules would
    //      otherwise force).
    for (int rb = wave; rb < NBLK; rb += NWAVES) {
        const int arow = (rb * 16 + nl) * KP;
        v2f aR[KSTEPS], aI[KSTEPS], aSv[KSTEPS];
        #pragma unroll
        for (int ks = 0; ks < KSTEPS; ++ks) {
            const int k0 = ks * 4 + half * 2;
            aR[ks]  = *(const v2f*)(lds + X0R + arow + k0);
            aI[ks]  = *(const v2f*)(lds + X0I + arow + k0);
            aSv[ks] = aR[ks] + aI[ks];
        }
        #pragma unroll
        for (int nt = 0; nt < 3; ++nt) {
            v8f t1 = {};   // Ra @ Rb
            v8f t2 = {};   // Ia @ Ib
            v8f t3 = {};   // (Ra+Ia) @ (Rb+Ib)
            #pragma unroll
            for (int ks = 0; ks < KSTEPS; ++ks) {
                const int k0 = ks * 4 + half * 2;
                const int bi = (k0 >> 1) * NP + nt * 16 + nl;
                const v2f bR = ((const v2f*)(lds + B1R))[bi];
                const v2f bI = ((const v2f*)(lds + B1I))[bi];
                const v2f bS = bR + bI;
                t1 = __builtin_amdgcn_wmma_f32_16x16x4_f32(false, aR[ks],  false, bR, (short)0, t1, false, false);
                t2 = __builtin_amdgcn_wmma_f32_16x16x4_f32(false, aI[ks],  false, bI, (short)0, t2, false, false);
                t3 = __builtin_amdgcn_wmma_f32_16x16x4_f32(false, aSv[ks], false, bS, (short)0, t3, false, false);
            }
            const int col = nt * 16 + nl;
            if (col < KP) {   // padded B columns >=33 produce zeros; cols >=36 dropped
                #pragma unroll
                for (int v = 0; v < 8; ++v) {
                    const float re = t1[v] - t2[v];
                    const float im = t3[v] - t1[v] - t2[v];
                    const int row = rb * 16 + half * 8 + v;
                    lds[PWR + row * KP + col] = re * re + im * im;
                }
            }
        }
    }
    __syncthreads();

    // ---- stage 2: Z = P @ F_sub^H. A is REAL -> only 2 WMMA per k-step.
    for (int rb = wave; rb < NBLK; rb += NWAVES) {
        const int arow = (rb * 16 + nl) * KP;
        v2f aP[KSTEPS];
        #pragma unroll
        for (int ks = 0; ks < KSTEPS; ++ks) {
            const int k0 = ks * 4 + half * 2;
            aP[ks] = *(const v2f*)(lds + PWR + arow + k0);
        }
        #pragma unroll
        for (int nt = 0; nt < 3; ++nt) {
            v8f zr = {};
            v8f zi = {};
            #pragma unroll
            for (int ks = 0; ks < KSTEPS; ++ks) {
                const int k0 = ks * 4 + half * 2;
                const int bi = (k0 >> 1) * NP + nt * 16 + nl;
                const v2f bR = ((const v2f*)(lds + B2R))[bi];
                const v2f bI = ((const v2f*)(lds + B2I))[bi];
                zr = __builtin_amdgcn_wmma_f32_16x16x4_f32(false, aP[ks], false, bR, (short)0, zr, false, false);
                zi = __builtin_amdgcn_wmma_f32_16x16x4_f32(false, aP[ks], false, bI, (short)0, zi, false, false);
            }
            const int col = nt * 16 + nl;
            if (col < KP) {
                #pragma unroll
                for (int v = 0; v < 8; ++v) {
                    const int row = rb * 16 + half * 8 + v;
                    lds[X0R + row * KP + col] = zr[v];
                    lds[X0I + row * KP + col] = zi[v];
                }
            }
        }
    }
    __syncthreads();

    // ---- inverse symbol DFT (F_sym^H, 5x5 complex) per (a, v) column,
    //      fused with the nontemporal complex64 store to HBM.
    const long outBase = (long)n * SLICE;
    for (int c = tid; c < NANT * NSUB; c += THREADS) {
        const int a = c / NSUB, v = c - a * NSUB;
        float zr[NSYM], zi[NSYM];
        #pragma unroll
        for (int t = 0; t < NSYM; ++t) {
            zr[t] = lds[X0R + (t * NANT + a) * KP + v];
            zi[t] = lds[X0I + (t * NANT + a) * KP + v];
        }
        #pragma unroll
        for (int s = 0; s < NSYM; ++s) {
            float yr = 0.f, yi = 0.f;
            #pragma unroll
            for (int t = 0; t < NSYM; ++t) {
                const float fr = lds[FSR + t * NSYM + s];
                const float fi = lds[FSI + t * NSYM + s];
                yr += fr * zr[t] + fi * zi[t];   // conj(F[t,s]) * Z[t]
                yi += fr * zi[t] - fi * zr[t];
            }
            const long o = outBase + (long)(s * NANT + a) * NSUB + v;
            v2f cv;
            cv.x = yr;
            cv.y = yi;
            __builtin_nontemporal_store(cv, (v2f*)(gout + 2 * o));
        }
    }
}

extern "C" void kernel_launch(void* const* d_in, const int* in_sizes, int n_in,
                              void* d_out, int out_size, void* d_ws, size_t ws_size,
                              hipStream_t stream) {
    const float* xr  = (const float*)d_in[0];
    const float* xi  = (const float*)d_in[1];
    const float* fsr = (const float*)d_in[2];
    const float* fsi = (const float*)d_in[3];
    const float* fbr = (const float*)d_in[4];
    const float* fbi = (const float*)d_in[5];
    float* out = (float*)d_out;

    const int N = in_sizes[0] / SLICE;           // 4096 batch elements
    ssa_autocorr_kernel<<<N, THREADS, 0, stream>>>(xr, xi, fsr, fsi, fbr, fbi, out);
}